// TransformerDecoderCross_4904852652786
// MI455X (gfx1250) — compile-verified
//
#include <hip/hip_runtime.h>
#include <hip/hip_bf16.h>
#include <math.h>

// ---------------------------------------------------------------------------
// Problem constants (from reference): B=4, N=512, K=256, D=128, H=4, hd=32, L=2
// ---------------------------------------------------------------------------
#define BB 4
#define NN 512
#define KK 256
#define DD 128
#define HH 4
#define HD 32
#define LL 2

typedef __attribute__((ext_vector_type(2))) float v2f;
typedef __attribute__((ext_vector_type(8))) float v8f;

// ---------------------------------------------------------------------------
// FP32 WMMA GEMM:  C[M,Nn] = A[M,Kd] @ W[Nn,Kd]^T (+ bias) (opt. relu)
// One wave32 computes one 32x32 macro-tile = 2x2 grid of 16x16 WMMA tiles.
// Per K-step of 4: 2 A-operand loads + 2 B-operand loads feed 4 WMMAs
// (2x operand reuse vs. a 16x16 tile -> 8 FLOP/byte from L2).
//
// ISA layouts (CDNA5 ISA), V_WMMA_F32_16X16X4_F32:
//   A 16x4 : lanes 0-15 hold M=lane (K=0 in vgpr0, K=1 in vgpr1),
//            lanes 16-31 hold M=lane-16 (K=2 in vgpr0, K=3 in vgpr1)
//   B 4x16 : column N = lane&15; lanes 0-15 hold K=0/1, lanes 16-31 hold K=2/3
//   C 16x16: vgpr r -> M = r + (lane<16 ? 0 : 8), N = lane&15
// ---------------------------------------------------------------------------
__global__ __launch_bounds__(128)
void gemm_wmma_f32(const float* __restrict__ A, int lda,
                   const float* __restrict__ W, int ldw,
                   const float* __restrict__ bias,
                   float* __restrict__ C, int ldc,
                   int M, int Nn, int Kd, int relu)
{
    const int wavesPerBlock = blockDim.x >> 5;
    const int tile  = blockIdx.x * wavesPerBlock + (threadIdx.x >> 5);
    const int tilesN = Nn >> 5;                  // 32-wide macro tiles
    const int tilesTotal = (M >> 5) * tilesN;
    if (tile >= tilesTotal) return;

    const int tm = tile / tilesN;
    const int tn = tile % tilesN;

    const int lane = threadIdx.x & 31;
    const int half = lane >> 4;      // 0 -> K pair {0,1}, 1 -> K pair {2,3}
    const int l16  = lane & 15;

    const float* __restrict__ arow0 = A + (size_t)(tm * 32 + l16) * lda + half * 2;
    const float* __restrict__ arow1 = arow0 + (size_t)16 * lda;
    const float* __restrict__ wrow0 = W + (size_t)(tn * 32 + l16) * ldw + half * 2;
    const float* __restrict__ wrow1 = wrow0 + (size_t)16 * ldw;

    v8f c00 = {}, c01 = {}, c10 = {}, c11 = {};
    for (int k0 = 0; k0 < Kd; k0 += 4) {
        v2f a0, a1, b0, b1;
        a0.x = arow0[k0]; a0.y = arow0[k0 + 1];
        a1.x = arow1[k0]; a1.y = arow1[k0 + 1];
        b0.x = wrow0[k0]; b0.y = wrow0[k0 + 1];
        b1.x = wrow1[k0]; b1.y = wrow1[k0 + 1];
        c00 = __builtin_amdgcn_wmma_f32_16x16x4_f32(false, a0, false, b0,
                                                    (short)0, c00, false, false);
        c01 = __builtin_amdgcn_wmma_f32_16x16x4_f32(false, a0, false, b1,
                                                    (short)0, c01, false, false);
        c10 = __builtin_amdgcn_wmma_f32_16x16x4_f32(false, a1, false, b0,
                                                    (short)0, c10, false, false);
        c11 = __builtin_amdgcn_wmma_f32_16x16x4_f32(false, a1, false, b1,
                                                    (short)0, c11, false, false);
    }

    const int n0 = tn * 32 + l16;        // column of sub-tiles j=0
    const int n1 = n0 + 16;              // column of sub-tiles j=1
    const float bn0 = bias ? bias[n0] : 0.0f;
    const float bn1 = bias ? bias[n1] : 0.0f;
    const int m0 = tm * 32 + half * 8;   // row base, sub-tiles i=0
    const int m1 = m0 + 16;              // row base, sub-tiles i=1

#pragma unroll
    for (int r = 0; r < 8; ++r) {
        float v00 = c00[r] + bn0;
        float v01 = c01[r] + bn1;
        float v10 = c10[r] + bn0;
        float v11 = c11[r] + bn1;
        if (relu) {
            v00 = fmaxf(v00, 0.0f); v01 = fmaxf(v01, 0.0f);
            v10 = fmaxf(v10, 0.0f); v11 = fmaxf(v11, 0.0f);
        }
        C[(size_t)(m0 + r) * ldc + n0] = v00;
        C[(size_t)(m0 + r) * ldc + n1] = v01;
        C[(size_t)(m1 + r) * ldc + n0] = v10;
        C[(size_t)(m1 + r) * ldc + n1] = v11;
    }
}

// ---------------------------------------------------------------------------
// Multi-head attention, hd=32, one block (256 threads) per (b, h, q) row.
// Q rows: (b*Sq + q) with stride ldq, head offset h*32.  K/V rows: (b*Sk + k).
// mask: (B, Sk) bool (uint8).  O: (B*Sq, 128) dense.
// ---------------------------------------------------------------------------
__global__ __launch_bounds__(256)
void attn_kernel(const float* __restrict__ Q, int ldq,
                 const float* __restrict__ Km, int ldk,
                 const float* __restrict__ V, int ldv,
                 const unsigned char* __restrict__ mask,
                 float* __restrict__ O,
                 int Sq, int Sk)
{
    __shared__ float s_q[HD];
    __shared__ float s_p[NN];      // up to 512 scores
    __shared__ float s_red[256];
    __shared__ float s_o[8 * HD];

    const int blk = blockIdx.x;
    const int q = blk % Sq;
    const int h = (blk / Sq) % HH;
    const int b = blk / (Sq * HH);
    const int t = threadIdx.x;

    const float* qrow = Q + (size_t)(b * Sq + q) * ldq + h * HD;
    if (t < HD) s_q[t] = qrow[t];
    __syncthreads();

    const float scale = 0.17677669529663687f;  // 1/sqrt(32)

    for (int k = t; k < Sk; k += 256) {
        const float* krow = Km + (size_t)(b * Sk + k) * ldk + h * HD;
        float s = 0.0f;
#pragma unroll
        for (int d = 0; d < HD; ++d) s += s_q[d] * krow[d];
        s *= scale;
        if (!mask[b * Sk + k]) s = -1000000000.0f;
        s_p[k] = s;
    }
    __syncthreads();

    // block max
    float m = -INFINITY;
    for (int k = t; k < Sk; k += 256) m = fmaxf(m, s_p[k]);
    s_red[t] = m;
    __syncthreads();
    for (int off = 128; off > 0; off >>= 1) {
        if (t < off) s_red[t] = fmaxf(s_red[t], s_red[t + off]);
        __syncthreads();
    }
    m = s_red[0];
    __syncthreads();

    // exp + block sum
    float sum = 0.0f;
    for (int k = t; k < Sk; k += 256) {
        float e = expf(s_p[k] - m);
        s_p[k] = e;
        sum += e;
    }
    s_red[t] = sum;
    __syncthreads();
    for (int off = 128; off > 0; off >>= 1) {
        if (t < off) s_red[t] += s_red[t + off];
        __syncthreads();
    }
    const float inv = 1.0f / s_red[0];
    __syncthreads();

    // weighted sum over V: thread t handles dim d = t&31, key-chunk t>>5 (of 8)
    const int d = t & (HD - 1);
    const int chunk = t >> 5;
    float acc = 0.0f;
    for (int k = chunk; k < Sk; k += 8) {
        acc += s_p[k] * V[(size_t)(b * Sk + k) * ldv + h * HD + d];
    }
    s_o[chunk * HD + d] = acc;
    __syncthreads();

    if (t < HD) {
        float o = 0.0f;
#pragma unroll
        for (int c = 0; c < 8; ++c) o += s_o[c * HD + t];
        O[(size_t)(b * Sq + q) * DD + h * HD + t] = o * inv;
    }
}

// ---------------------------------------------------------------------------
// X = LayerNorm(X + Delta) * w + b     (one 128-thread block per row of D=128)
// ---------------------------------------------------------------------------
__global__ __launch_bounds__(128)
void add_ln_kernel(float* __restrict__ X, const float* __restrict__ Dl,
                   const float* __restrict__ w, const float* __restrict__ b)
{
    __shared__ float red[DD];
    const int row = blockIdx.x;
    const int t = threadIdx.x;
    const size_t base = (size_t)row * DD;

    float v = X[base + t] + Dl[base + t];

    red[t] = v;
    __syncthreads();
    for (int off = 64; off > 0; off >>= 1) {
        if (t < off) red[t] += red[t + off];
        __syncthreads();
    }
    const float mean = red[0] * (1.0f / DD);
    __syncthreads();

    const float dv = v - mean;
    red[t] = dv * dv;
    __syncthreads();
    for (int off = 64; off > 0; off >>= 1) {
        if (t < off) red[t] += red[t + off];
        __syncthreads();
    }
    const float var = red[0] * (1.0f / DD);

    X[base + t] = dv * rsqrtf(var + 1e-5f) * w[t] + b[t];
}

// ---------------------------------------------------------------------------
// X = concat(mask_nodes[B,N,2], sincos_pe[B,N,8]) @ Wp[128,10]^T + bp
// ---------------------------------------------------------------------------
__global__ __launch_bounds__(256)
void input_proj_kernel(const float* __restrict__ mn, const float* __restrict__ pe,
                       const float* __restrict__ Wp, const float* __restrict__ bp,
                       float* __restrict__ X)
{
    const int idx = blockIdx.x * 256 + threadIdx.x;
    if (idx >= BB * NN * DD) return;
    const int row = idx >> 7;
    const int o = idx & (DD - 1);
    const float* wr = Wp + o * 10;
    const float* mnr = mn + row * 2;
    const float* per = pe + row * 8;
    float acc = bp[o] + wr[0] * mnr[0] + wr[1] * mnr[1];
#pragma unroll
    for (int j = 0; j < 8; ++j) acc += wr[2 + j] * per[j];
    X[idx] = acc;
}

// ---------------------------------------------------------------------------
// Edge head: out[b,i,j] = sum_d relu(u[b,i,d]+v[b,j,d]+b1[d]) * w2[d] + b2,
// diagonal forced to -inf. One block (256 thr) per (b,i).
// ---------------------------------------------------------------------------
__global__ __launch_bounds__(256)
void edge_kernel(const float* __restrict__ U, const float* __restrict__ Vv,
                 const float* __restrict__ b1, const float* __restrict__ w2,
                 const float* __restrict__ b2p, float* __restrict__ Out)
{
    __shared__ float s_u[DD];
    __shared__ float s_w[DD];
    const int bi = blockIdx.x;            // bi = b*N + i
    const int b = bi >> 9;
    const int i = bi & (NN - 1);
    const int t = threadIdx.x;

    if (t < DD) {
        s_u[t] = U[(size_t)bi * DD + t] + b1[t];
        s_w[t] = w2[t];
    }
    __syncthreads();

    const float b2 = b2p[0];
    for (int j = t; j < NN; j += 256) {
        const float* vr = Vv + ((size_t)b * NN + j) * DD;
        if (j + 256 < NN)
            __builtin_prefetch(Vv + ((size_t)b * NN + j + 256) * DD, 0, 3);
        float acc = 0.0f;
#pragma unroll 4
        for (int d = 0; d < DD; ++d) acc += fmaxf(s_u[d] + vr[d], 0.0f) * s_w[d];
        float val = acc + b2;
        if (j == i) val = -INFINITY;
        Out[((size_t)b * NN + i) * NN + j] = val;
    }
}

// ---------------------------------------------------------------------------
static void launch_gemm(const float* A, int lda, const float* W, int ldw,
                        const float* bias, float* C, int ldc,
                        int M, int Nn, int Kd, int relu, hipStream_t s)
{
    const int tiles = (M / 32) * (Nn / 32);  // 32x32 macro-tiles, one per wave
    const int wpb = 4;                       // 4 waves (tiles) per block
    const int blocks = (tiles + wpb - 1) / wpb;
    gemm_wmma_f32<<<blocks, 32 * wpb, 0, s>>>(A, lda, W, ldw, bias, C, ldc,
                                              M, Nn, Kd, relu);
}

extern "C" void kernel_launch(void* const* d_in, const int* in_sizes, int n_in,
                              void* d_out, int out_size, void* d_ws, size_t ws_size,
                              hipStream_t stream)
{
    const float* mask_nodes   = (const float*)d_in[0];
    const float* sincos_pe    = (const float*)d_in[1];
    const float* h_pool       = (const float*)d_in[2];
    const float* input_proj_w = (const float*)d_in[3];
    const float* input_proj_b = (const float*)d_in[4];
    const float* sa_qkv_w     = (const float*)d_in[5];
    const float* sa_qkv_b     = (const float*)d_in[6];
    const float* sa_out_w     = (const float*)d_in[7];
    const float* sa_out_b     = (const float*)d_in[8];
    const float* ca_qkv_w     = (const float*)d_in[9];
    const float* ca_qkv_b     = (const float*)d_in[10];
    const float* ca_out_w     = (const float*)d_in[11];
    const float* ca_out_b     = (const float*)d_in[12];
    const float* lin1_w       = (const float*)d_in[13];
    const float* lin1_b       = (const float*)d_in[14];
    const float* lin2_w       = (const float*)d_in[15];
    const float* lin2_b       = (const float*)d_in[16];
    const float* ln1_w        = (const float*)d_in[17];
    const float* ln1_b        = (const float*)d_in[18];
    const float* ln2_w        = (const float*)d_in[19];
    const float* ln2_b        = (const float*)d_in[20];
    const float* ln3_w        = (const float*)d_in[21];
    const float* ln3_b        = (const float*)d_in[22];
    const float* edge_w1      = (const float*)d_in[23];
    const float* edge_b1      = (const float*)d_in[24];
    const float* edge_w2      = (const float*)d_in[25];
    const float* edge_b2      = (const float*)d_in[26];
    const unsigned char* node_mask = (const unsigned char*)d_in[27];
    const unsigned char* kv_mask   = (const unsigned char*)d_in[28];

    // workspace layout (floats)
    float* ws = (float*)d_ws;
    const size_t SZ_X   = (size_t)BB * NN * DD;          // 262144
    const size_t SZ_QKV = (size_t)BB * NN * 3 * DD;      // 786432
    float* X     = ws;
    float* QKV   = X + SZ_X;                             // scratch (qkv / ffn / u+v)
    float* O     = QKV + SZ_QKV;                         // attention output
    float* DELTA = O + SZ_X;                             // residual delta

    const int Mx = BB * NN;      // 2048 rows of x
    const int Mh = BB * KK;      // 1024 rows of h_pool

    // x = input projection
    input_proj_kernel<<<(Mx * DD + 255) / 256, 256, 0, stream>>>(
        mask_nodes, sincos_pe, input_proj_w, input_proj_b, X);

    for (int l = 0; l < LL; ++l) {
        // ---- self attention ----
        launch_gemm(X, DD, sa_qkv_w + (size_t)l * 3 * DD * DD, DD,
                    sa_qkv_b + (size_t)l * 3 * DD,
                    QKV, 3 * DD, Mx, 3 * DD, DD, 0, stream);

        attn_kernel<<<BB * HH * NN, 256, 0, stream>>>(
            QKV, 3 * DD,                 // Q at cols [0,128)
            QKV + DD, 3 * DD,            // K at cols [128,256)
            QKV + 2 * DD, 3 * DD,        // V at cols [256,384)
            node_mask, O, NN, NN);

        launch_gemm(O, DD, sa_out_w + (size_t)l * DD * DD, DD,
                    sa_out_b + (size_t)l * DD,
                    DELTA, DD, Mx, DD, DD, 0, stream);

        add_ln_kernel<<<Mx, DD, 0, stream>>>(X, DELTA,
                                             ln1_w + l * DD, ln1_b + l * DD);

        // ---- cross attention ----
        launch_gemm(X, DD, ca_qkv_w + (size_t)l * 3 * DD * DD, DD,
                    ca_qkv_b + (size_t)l * 3 * DD,
                    QKV, DD, Mx, DD, DD, 0, stream);                 // cross Q

        launch_gemm(h_pool, DD,
                    ca_qkv_w + (size_t)l * 3 * DD * DD + (size_t)DD * DD, DD,
                    ca_qkv_b + (size_t)l * 3 * DD + DD,
                    QKV + SZ_X, 2 * DD, Mh, 2 * DD, DD, 0, stream);  // cross K|V

        attn_kernel<<<BB * HH * NN, 256, 0, stream>>>(
            QKV, DD,                     // Q (2048 x 128)
            QKV + SZ_X, 2 * DD,          // K at cols [0,128) of kv buffer
            QKV + SZ_X + DD, 2 * DD,     // V at cols [128,256)
            kv_mask, O, NN, KK);

        launch_gemm(O, DD, ca_out_w + (size_t)l * DD * DD, DD,
                    ca_out_b + (size_t)l * DD,
                    DELTA, DD, Mx, DD, DD, 0, stream);

        add_ln_kernel<<<Mx, DD, 0, stream>>>(X, DELTA,
                                             ln2_w + l * DD, ln2_b + l * DD);

        // ---- feed-forward ----
        launch_gemm(X, DD, lin1_w + (size_t)l * 2 * DD * DD, DD,
                    lin1_b + (size_t)l * 2 * DD,
                    QKV, 2 * DD, Mx, 2 * DD, DD, 1, stream);         // relu fused

        launch_gemm(QKV, 2 * DD, lin2_w + (size_t)l * DD * 2 * DD, 2 * DD,
                    lin2_b + (size_t)l * DD,
                    DELTA, DD, Mx, DD, 2 * DD, 0, stream);

        add_ln_kernel<<<Mx, DD, 0, stream>>>(X, DELTA,
                                             ln3_w + l * DD, ln3_b + l * DD);
    }

    // ---- edge head ----
    // u = x @ edge_w1[:, :D]^T  (row stride 2D), v = x @ edge_w1[:, D:]^T
    float* U = QKV;
    float* Vv = QKV + SZ_X;
    launch_gemm(X, DD, edge_w1, 2 * DD, nullptr, U, DD, Mx, DD, DD, 0, stream);
    launch_gemm(X, DD, edge_w1 + DD, 2 * DD, nullptr, Vv, DD, Mx, DD, DD, 0, stream);

    edge_kernel<<<BB * NN, 256, 0, stream>>>(U, Vv, edge_b1, edge_w2, edge_b2,
                                             (float*)d_out);
}